// Classifier_78546361909690
// MI455X (gfx1250) — compile-verified
//
#include <hip/hip_runtime.h>
#include <hip/hip_bf16.h>

// Problem constants (match reference)
constexpr int N_NODES = 50000;
constexpr int N_EDGES = 800000;
constexpr int N_GRAPH = 128;
constexpr int HID     = 128;      // == IN_DIM
constexpr int KDIM    = 3 * HID;  // 384 (concat of 3 hop features)
constexpr int CLASSES = 10;

typedef __attribute__((ext_vector_type(2))) float v2f;
typedef __attribute__((ext_vector_type(8))) float v8f;

// ---------------------------------------------------------------- utilities
__device__ __forceinline__ void atomic_add_f32(float* p, float v) {
  // hardware global_atomic_add_f32 (device scope), no CAS loop
  unsafeAtomicAdd(p, v);
}

__global__ void zero_kernel(float* __restrict__ p, long n) {
  long i = (long)blockIdx.x * blockDim.x + threadIdx.x;
  long stride = (long)gridDim.x * blockDim.x;
  for (; i < n; i += stride) p[i] = 0.0f;
}

// ------------------------------------------------------------- degree/norm
__global__ void degree_kernel(const int* __restrict__ dst,
                              float* __restrict__ deg, int e) {
  int i = blockIdx.x * blockDim.x + threadIdx.x;
  if (i < e) atomic_add_f32(&deg[dst[i]], 1.0f);
}

__global__ void norm_kernel(float* __restrict__ deg_inout, int n) {
  int i = blockIdx.x * blockDim.x + threadIdx.x;
  if (i < n) {
    float d = deg_inout[i];
    d = (d < 1.0f) ? 1.0f : d;
    deg_inout[i] = rsqrtf(d);   // deg^(-1/2)
  }
}

// --------------------------------------------------------- edge aggregation
// One wave per edge; lane handles 4 features (float4).
// Fout[dst] += norm[src]*norm[dst] * Fin[src]   (fuses both norm scalings)
__global__ void edge_aggregate_kernel(const float* __restrict__ Fin,
                                      const int* __restrict__ src,
                                      const int* __restrict__ dst,
                                      const float* __restrict__ norm,
                                      float* __restrict__ Fout, int e) {
  int wave = (int)(((long)blockIdx.x * blockDim.x + threadIdx.x) >> 5);
  int lane = threadIdx.x & 31;
  if (wave >= e) return;
  int s = src[wave];
  int d = dst[wave];
  float c = norm[s] * norm[d];
  const float4 v = ((const float4*)(Fin + (long)s * HID))[lane];
  float* o = Fout + (long)d * HID + 4 * lane;
  atomic_add_f32(o + 0, c * v.x);
  atomic_add_f32(o + 1, c * v.y);
  atomic_add_f32(o + 2, c * v.z);
  atomic_add_f32(o + 3, c * v.w);
}

// -------------------------------------------------- fused concat-GEMM+ReLU
// Hout(N,128) = relu( concat(F0,F1,F2)(N,384) @ W(384,128) + b )
// One wave32 computes one 16x16 output tile via V_WMMA_F32_16X16X4_F32.
// A layout (ISA 7.12.2, 32-bit A 16x4): lane l -> row m=l&15, K-group kg=l>>4,
//   VGPR0/1 = A[m][2*kg+0 / +1]  -> one contiguous float2 load per step.
// B (4x16) mirrored over N: VGPR0/1 = B[2*kg+0 / +1][n].
// C/D: VGPR v, lane l -> D[v + 8*kg][n], n = l&15.
__global__ void __launch_bounds__(128)
tag_gemm_relu_kernel(const float* __restrict__ F0,
                     const float* __restrict__ F1,
                     const float* __restrict__ F2,
                     const float* __restrict__ W,   // (384,128) row-major
                     const float* __restrict__ b,   // (128)
                     float* __restrict__ Hout) {    // (N,128)
  const int lane = threadIdx.x & 31;
  const int waveId = blockIdx.x * 4 + (threadIdx.x >> 5);
  const int rt = waveId >> 3;              // row tile  (0..3124)
  const int ct = waveId & 7;               // col tile  (0..7)
  if (rt >= N_NODES / 16) return;          // wave-uniform -> EXEC stays all-1s

  const int m   = lane & 15;
  const int kg  = lane >> 4;               // K-group 0/1
  const int row = rt * 16 + m;
  const int col = ct * 16 + m;

  v8f acc;
  const float bv = b[col];
#pragma unroll
  for (int i = 0; i < 8; ++i) acc[i] = bv;

  const float* Fs[3] = {F0, F1, F2};
#pragma unroll 1
  for (int arr = 0; arr < 3; ++arr) {
    const float* __restrict__ F = Fs[arr];
    const float* __restrict__ arow = F + (long)row * HID + 2 * kg;
#pragma unroll 4
    for (int k = 0; k < HID; k += 4) {
      v2f a = *(const v2f*)(arow + k);
      const int krow = arr * HID + k + 2 * kg;
      v2f bb;
      bb[0] = W[(long)krow * HID + col];
      bb[1] = W[(long)(krow + 1) * HID + col];
      acc = __builtin_amdgcn_wmma_f32_16x16x4_f32(
          /*neg_a=*/false, a, /*neg_b=*/false, bb,
          /*c_mod=*/(short)0, acc, /*reuse_a=*/false, /*reuse_b=*/false);
    }
  }

#pragma unroll
  for (int v = 0; v < 8; ++v) {
    float x = acc[v];
    x = (x > 0.0f) ? x : 0.0f;                 // ReLU
    Hout[(long)(rt * 16 + v + 8 * kg) * HID + col] = x;
  }
}

// --------------------------------------------------------------- pooling
// One wave per node; lane covers 4 features.
__global__ void pool_kernel(const float* __restrict__ H,
                            const int* __restrict__ gid,
                            float* __restrict__ sums,   // (G,128)
                            float* __restrict__ cnts,   // (G)
                            int n) {
  int wave = (int)(((long)blockIdx.x * blockDim.x + threadIdx.x) >> 5);
  int lane = threadIdx.x & 31;
  if (wave >= n) return;
  int g = gid[wave];
  const float4 v = ((const float4*)(H + (long)wave * HID))[lane];
  float* o = sums + (long)g * HID + 4 * lane;
  atomic_add_f32(o + 0, v.x);
  atomic_add_f32(o + 1, v.y);
  atomic_add_f32(o + 2, v.z);
  atomic_add_f32(o + 3, v.w);
  if (lane == 0) atomic_add_f32(&cnts[g], 1.0f);
}

// ------------------------------------------------------------- classifier
__global__ void classifier_kernel(const float* __restrict__ sums,
                                  const float* __restrict__ cnts,
                                  const float* __restrict__ Wc,  // (128,10)
                                  const float* __restrict__ bc,  // (10)
                                  float* __restrict__ out) {     // (G,10)
  int i = blockIdx.x * blockDim.x + threadIdx.x;
  if (i >= N_GRAPH * CLASSES) return;
  int g = i / CLASSES;
  int c = i % CLASSES;
  float cnt = cnts[g];
  cnt = (cnt < 1.0f) ? 1.0f : cnt;
  float inv = 1.0f / cnt;
  float acc = bc[c];
#pragma unroll 8
  for (int k = 0; k < HID; ++k)
    acc = fmaf(sums[(long)g * HID + k] * inv, Wc[(long)k * CLASSES + c], acc);
  out[i] = acc;
}

// ================================================================ launcher
extern "C" void kernel_launch(void* const* d_in, const int* in_sizes, int n_in,
                              void* d_out, int out_size, void* d_ws, size_t ws_size,
                              hipStream_t stream) {
  const float* x    = (const float*)d_in[0];   // (N,128)
  const int*   src  = (const int*)  d_in[1];   // (E)
  const int*   dst  = (const int*)  d_in[2];   // (E)
  const int*   gid  = (const int*)  d_in[3];   // (N)
  const float* Wl[3] = {(const float*)d_in[4], (const float*)d_in[6], (const float*)d_in[8]};
  const float* bl[3] = {(const float*)d_in[5], (const float*)d_in[7], (const float*)d_in[9]};
  const float* Wc = (const float*)d_in[10];    // (128,10)
  const float* bc = (const float*)d_in[11];    // (10)
  float* out = (float*)d_out;                  // (G,10)

  // workspace layout (floats)
  const long NF = (long)N_NODES * HID;         // 6,400,000
  float* ws    = (float*)d_ws;
  float* norm  = ws;                           // N          (deg -> norm in place)
  float* A1    = ws + N_NODES;                 // N*128
  float* A2    = A1 + NF;                      // N*128
  float* B1    = A2 + NF;                      // N*128
  float* B2    = B1 + NF;                      // N*128
  float* sums  = B2 + NF;                      // G*128
  float* cnts  = sums + (long)N_GRAPH * HID;   // G
  const size_t need = (size_t)(cnts + N_GRAPH - ws) * sizeof(float);
  if (ws_size < need) return;

  const int ZB = 512, ZT = 256;

  // degree -> norm
  zero_kernel<<<ZB, ZT, 0, stream>>>(norm, N_NODES);
  degree_kernel<<<(N_EDGES + 255) / 256, 256, 0, stream>>>(dst, norm, N_EDGES);
  norm_kernel<<<(N_NODES + 255) / 256, 256, 0, stream>>>(norm, N_NODES);

  const int aggBlocks  = (N_EDGES * 32) / 256;       // one wave per edge
  const int gemmBlocks = (N_NODES / 16) * 8 / 4;     // 6250: 4 tiles/block

  const float* h_in = x;
  float* h_out = B1;
  for (int l = 0; l < 3; ++l) {
    zero_kernel<<<ZB, ZT, 0, stream>>>(A1, NF);
    edge_aggregate_kernel<<<aggBlocks, 256, 0, stream>>>(h_in, src, dst, norm, A1, N_EDGES);
    zero_kernel<<<ZB, ZT, 0, stream>>>(A2, NF);
    edge_aggregate_kernel<<<aggBlocks, 256, 0, stream>>>(A1, src, dst, norm, A2, N_EDGES);
    tag_gemm_relu_kernel<<<gemmBlocks, 128, 0, stream>>>(h_in, A1, A2, Wl[l], bl[l], h_out);
    h_in  = h_out;
    h_out = (h_out == B1) ? B2 : B1;
  }
  // h_in now points at final node features

  zero_kernel<<<ZB, ZT, 0, stream>>>(sums, (long)N_GRAPH * HID + N_GRAPH);
  pool_kernel<<<(N_NODES * 32) / 256, 256, 0, stream>>>(h_in, gid, sums, cnts, N_NODES);
  classifier_kernel<<<(N_GRAPH * CLASSES + 255) / 256, 256, 0, stream>>>(sums, cnts, Wc, bc, out);
}